// ScatterADPhysics_5437428597337
// MI455X (gfx1250) — compile-verified
//
#include <hip/hip_runtime.h>
#include <math.h>

typedef float v2f __attribute__((ext_vector_type(2)));
typedef float v8f __attribute__((ext_vector_type(8)));

#define M_TOT 2048   // B*L
#define L_SEQ 512
#define HIDN  64
#define HIN   128

// workspace layout (floats)
constexpr int OFF_ZT   = 0;
constexpr int OFF_WH   = OFF_ZT   + M_TOT*HIDN;
constexpr int OFF_SISJ = OFF_WH   + M_TOT*HIDN;
constexpr int OFF_ZG   = OFF_SISJ + M_TOT*HIDN;
constexpr int OFF_Z    = OFF_ZG   + M_TOT*HIDN;
constexpr int OFF_ZN   = OFF_Z    + M_TOT*HIDN;
constexpr int OFF_QN   = OFF_ZN   + M_TOT*HIDN;
constexpr int OFF_KN   = OFF_QN   + M_TOT*HIDN;
constexpr int OFF_E1   = OFF_KN   + M_TOT*HIDN;
constexpr int OFF_E2   = OFF_E1   + M_TOT;
constexpr int OFF_ACC  = OFF_E2   + M_TOT;      // zeroed region begins
constexpr int OFF_RS   = OFF_ACC;               // row_sum [2048]
constexpr int OFF_CS   = OFF_RS   + M_TOT;      // col_sum [2048]
constexpr int OFF_DIAG = OFF_CS   + M_TOT;      // diag    [2048]
constexpr int OFF_MEAN = OFF_DIAG + M_TOT;      // mean_acc[64]
constexpr int OFF_TACC = OFF_MEAN + HIDN;       // time_acc[1]
constexpr int ACC_N    = 3*M_TOT + HIDN + 1;

__device__ inline v8f wmma4(v2f a, v2f b, v8f c) {
  return __builtin_amdgcn_wmma_f32_16x16x4_f32(false, a, false, b, (short)0, c, false, false);
}
__device__ inline float wave_sum32(float v) {
  for (int m = 1; m < 32; m <<= 1) v += __shfl_xor(v, m, 32);
  return v;
}
__device__ inline float half_sum16(float v) {
  for (int m = 1; m < 16; m <<= 1) v += __shfl_xor(v, m, 32);
  return v;
}

__global__ void k_zero(float* p, int n) {
  int i = blockIdx.x * blockDim.x + threadIdx.x;
  if (i < n) p[i] = 0.f;
}

// z_t = relu(h @ W_in + b_in)   [2048,128]@[128,64]
__global__ void k_gemm_zt(const float* __restrict__ h, const float* __restrict__ W_in,
                          const float* __restrict__ b_in, float* __restrict__ zt) {
  int wave = (blockIdx.x * blockDim.x + threadIdx.x) >> 5;
  int lane = threadIdx.x & 31, hf = lane >> 4, lm = lane & 15;
  if (wave >= (M_TOT/16)*(HIDN/16)) return;
  int m0 = (wave >> 2) * 16, n0 = (wave & 3) * 16;
  v8f acc = {};
  for (int k0 = 0; k0 < HIN; k0 += 4) {
    int ka = k0 + 2*hf;
    v2f a, b;
    a[0] = h[(m0+lm)*HIN + ka];     a[1] = h[(m0+lm)*HIN + ka + 1];
    b[0] = W_in[ka*HIDN + n0+lm];   b[1] = W_in[(ka+1)*HIDN + n0+lm];
    acc = wmma4(a, b, acc);
  }
  for (int r = 0; r < 8; ++r) {
    int row = m0 + r + 8*hf, col = n0 + lm;
    float v = acc[r] + b_in[col];
    zt[row*HIDN + col] = v > 0.f ? v : 0.f;
  }
}

// Wh = z_t @ Wg   [2048,64]@[64,64]
__global__ void k_gemm_wh(const float* __restrict__ zt, const float* __restrict__ Wg,
                          float* __restrict__ Wh) {
  int wave = (blockIdx.x * blockDim.x + threadIdx.x) >> 5;
  int lane = threadIdx.x & 31, hf = lane >> 4, lm = lane & 15;
  if (wave >= (M_TOT/16)*(HIDN/16)) return;
  int m0 = (wave >> 2) * 16, n0 = (wave & 3) * 16;
  v8f acc = {};
  for (int k0 = 0; k0 < HIDN; k0 += 4) {
    int ka = k0 + 2*hf;
    v2f a, b;
    a[0] = zt[(m0+lm)*HIDN + ka];  a[1] = zt[(m0+lm)*HIDN + ka + 1];
    b[0] = Wg[ka*HIDN + n0+lm];    b[1] = Wg[(ka+1)*HIDN + n0+lm];
    acc = wmma4(a, b, acc);
  }
  for (int r = 0; r < 8; ++r)
    Wh[(m0 + r + 8*hf)*HIDN + n0 + lm] = acc[r];
}

// si_sj[:, 0:32]=x_raw@Wi, [:,32:64]=x_raw@Wj  (Wi=Wm1[0:16], Wj=Wm1[16:32], 32 cols each)
__global__ void k_gemm_sisj(const float* __restrict__ xr, const float* __restrict__ Wm1,
                            float* __restrict__ sisj) {
  int wave = (blockIdx.x * blockDim.x + threadIdx.x) >> 5;
  int lane = threadIdx.x & 31, hf = lane >> 4, lm = lane & 15;
  if (wave >= (M_TOT/16)*(HIDN/16)) return;
  int m0 = (wave >> 2) * 16, n0 = (wave & 3) * 16;
  int n = n0 + lm;
  v8f acc = {};
  for (int k0 = 0; k0 < 16; k0 += 4) {
    int ka = k0 + 2*hf;
    v2f a, b;
    a[0] = xr[(m0+lm)*16 + ka];  a[1] = xr[(m0+lm)*16 + ka + 1];
    if (n < 32) { b[0] = Wm1[ka*32 + n];          b[1] = Wm1[(ka+1)*32 + n]; }
    else        { b[0] = Wm1[(16+ka)*32 + n-32];  b[1] = Wm1[(17+ka)*32 + n-32]; }
    acc = wmma4(a, b, acc);
  }
  for (int r = 0; r < 8; ++r)
    sisj[(m0 + r + 8*hf)*HIDN + n] = acc[r];
}

// e1 = Wh@a1, e2 = Wh@a2 per row
__global__ void k_rowdots(const float* __restrict__ Wh, const float* __restrict__ a1,
                          const float* __restrict__ a2, float* __restrict__ e1,
                          float* __restrict__ e2) {
  int wave = (blockIdx.x * blockDim.x + threadIdx.x) >> 5;
  int lane = threadIdx.x & 31;
  if (wave >= M_TOT) return;
  float v0 = Wh[wave*HIDN + lane], v1 = Wh[wave*HIDN + 32 + lane];
  float s1 = wave_sum32(v0*a1[lane] + v1*a1[32+lane]);
  float s2 = wave_sum32(v0*a2[lane] + v1*a2[32+lane]);
  if (lane == 0) { e1[wave] = s1; e2[wave] = s2; }
}

// banded attention: bias MLP (32 lanes = 32 hidden dims), softmax, z_g = elu(attn@Wh)
__global__ void k_attn(const float* __restrict__ sisj, const float* __restrict__ dt_emb,
                       const float* __restrict__ Wm1, const float* __restrict__ bm1,
                       const float* __restrict__ Wm2, const float* __restrict__ bm2,
                       const float* __restrict__ e1, const float* __restrict__ e2,
                       const float* __restrict__ Wh, float* __restrict__ zg) {
  int wave = (blockIdx.x * blockDim.x + threadIdx.x) >> 5;
  int lane = threadIdx.x & 31;
  if (wave >= M_TOT) return;
  int b = wave >> 9, i = wave & (L_SEQ - 1);
  float sd[4];
  for (int d = 0; d < 4; ++d) {
    float s = 0.f;
    for (int e = 0; e < 8; ++e) s += dt_emb[d*8 + e] * Wm1[(32 + e)*32 + lane];
    sd[d] = s;
  }
  float si_v = sisj[wave*HIDN + lane];
  float bm1v = bm1[lane], wm2v = Wm2[lane], bm2v = bm2[0];
  float e1i = e1[wave];
  int j0 = i - 3 < 0 ? 0 : i - 3;
  int j1 = i + 3 > L_SEQ - 1 ? L_SEQ - 1 : i + 3;
  int cnt = j1 - j0 + 1;
  float ev[7];
  for (int j = j0; j <= j1; ++j) {
    int mj = (b << 9) + j;
    int dt = i > j ? (i - j) : 0;
    float hid = si_v + sisj[mj*HIDN + 32 + lane] + sd[dt] + bm1v;
    hid = hid > 0.f ? hid : 0.f;
    float bias = wave_sum32(hid * wm2v) + bm2v;
    float x = e1i + e2[mj];
    float le = x >= 0.f ? x : 0.2f * x;
    ev[j - j0] = le + bias - 0.2f * (float)dt;
  }
  float mx = -1e30f;
  for (int t = 0; t < cnt; ++t) mx = fmaxf(mx, ev[t]);
  float s = 0.f;
  for (int t = 0; t < cnt; ++t) { ev[t] = __expf(ev[t] - mx); s += ev[t]; }
  float inv = 1.f / s;
  float a0 = 0.f, a1v = 0.f;
  for (int t = 0; t < cnt; ++t) {
    int mj = (b << 9) + j0 + t;
    float w = ev[t] * inv;
    a0  += w * Wh[mj*HIDN + lane];
    a1v += w * Wh[mj*HIDN + 32 + lane];
  }
  zg[wave*HIDN + lane]      = a0  > 0.f ? a0  : expm1f(a0);
  zg[wave*HIDN + 32 + lane] = a1v > 0.f ? a1v : expm1f(a1v);
}

// z = relu([z_t|z_g] @ Wf + bf)   K=128
__global__ void k_gemm_z(const float* __restrict__ zt, const float* __restrict__ zg,
                         const float* __restrict__ Wf, const float* __restrict__ bf,
                         float* __restrict__ z) {
  int wave = (blockIdx.x * blockDim.x + threadIdx.x) >> 5;
  int lane = threadIdx.x & 31, hf = lane >> 4, lm = lane & 15;
  if (wave >= (M_TOT/16)*(HIDN/16)) return;
  int m0 = (wave >> 2) * 16, n0 = (wave & 3) * 16;
  v8f acc = {};
  for (int k0 = 0; k0 < 2*HIDN; k0 += 4) {
    int ka = k0 + 2*hf;
    v2f a, b;
    const float* A = (ka < HIDN) ? zt : zg;
    int kk = (ka < HIDN) ? ka : ka - HIDN;
    a[0] = A[(m0+lm)*HIDN + kk];  a[1] = A[(m0+lm)*HIDN + kk + 1];
    b[0] = Wf[ka*HIDN + n0+lm];   b[1] = Wf[(ka+1)*HIDN + n0+lm];
    acc = wmma4(a, b, acc);
  }
  for (int r = 0; r < 8; ++r) {
    int row = m0 + r + 8*hf, col = n0 + lm;
    float v = acc[r] + bf[col];
    z[row*HIDN + col] = v > 0.f ? v : 0.f;
  }
}

// per-row normalize z->zn, z_t->qn, z_g->kn; accumulate mean(zn)
__global__ void k_rowstats(const float* __restrict__ z, const float* __restrict__ zt,
                           const float* __restrict__ zg, float* __restrict__ zn,
                           float* __restrict__ qn, float* __restrict__ kn,
                           float* __restrict__ mean_acc) {
  int wave = (blockIdx.x * blockDim.x + threadIdx.x) >> 5;
  int lane = threadIdx.x & 31;
  if (wave >= M_TOT) return;
  const float* src[3] = { z, zt, zg };
  float* dst[3] = { zn, qn, kn };
  for (int t = 0; t < 3; ++t) {
    float v0 = src[t][wave*HIDN + lane], v1 = src[t][wave*HIDN + 32 + lane];
    float n2 = wave_sum32(v0*v0 + v1*v1);
    float inv = 1.f / fmaxf(sqrtf(n2), 1e-8f);
    float o0 = v0 * inv, o1 = v1 * inv;
    dst[t][wave*HIDN + lane] = o0;
    dst[t][wave*HIDN + 32 + lane] = o1;
    if (t == 0) {
      atomicAdd(&mean_acc[lane], o0);
      atomicAdd(&mean_acc[32 + lane], o1);
    }
  }
}

// l_time accumulation: (1 - cos(zn[b,i+1], zn[b,i]))^2
__global__ void k_time(const float* __restrict__ zn, float* __restrict__ tacc) {
  int wave = (blockIdx.x * blockDim.x + threadIdx.x) >> 5;
  int lane = threadIdx.x & 31;
  if (wave >= M_TOT) return;
  int i = wave & (L_SEQ - 1);
  if (i == L_SEQ - 1) return;
  float v = zn[wave*HIDN + lane]      * zn[(wave+1)*HIDN + lane] +
            zn[wave*HIDN + 32 + lane] * zn[(wave+1)*HIDN + 32 + lane];
  float c = wave_sum32(v);
  if (lane == 0) { float d = 1.f - c; atomicAdd(tacc, d*d); }
}

// logits = qn @ kn^T / 0.1 ; fold exp into row/col sums, capture diagonal
__global__ void k_logits(const float* __restrict__ qn, const float* __restrict__ kn,
                         float* __restrict__ row_sum, float* __restrict__ col_sum,
                         float* __restrict__ diag) {
  int wave = (blockIdx.x * blockDim.x + threadIdx.x) >> 5;
  int lane = threadIdx.x & 31, hf = lane >> 4, lm = lane & 15;
  if (wave >= (M_TOT/16)*(M_TOT/16)) return;
  int m0 = (wave >> 7) * 16, n0 = (wave & 127) * 16;
  v8f acc = {};
  for (int k0 = 0; k0 < HIDN; k0 += 4) {
    int ka = k0 + 2*hf;
    v2f a, b;
    a[0] = qn[(m0+lm)*HIDN + ka];  a[1] = qn[(m0+lm)*HIDN + ka + 1];
    b[0] = kn[(n0+lm)*HIDN + ka];  b[1] = kn[(n0+lm)*HIDN + ka + 1];  // B = kn^T
    acc = wmma4(a, b, acc);
  }
  float colpart = 0.f;
  for (int r = 0; r < 8; ++r) {
    int row = m0 + r + 8*hf, col = n0 + lm;
    float l = acc[r] * 10.f;                 // / TEMP
    if (row == col) diag[row] = l;
    float ex = __expf(l);                    // logits bounded by ~10: stable
    colpart += ex;
    float rs = half_sum16(ex);
    if (lm == 0) atomicAdd(&row_sum[row], rs);
  }
  float ct = colpart + __shfl_xor(colpart, 16, 32);
  if (hf == 0) atomicAdd(&col_sum[n0 + lm], ct);
}

// final scalar reductions -> out[4] = {loss, l_contrast, l_scatter, l_time}
__global__ void k_final(const float* __restrict__ row_sum, const float* __restrict__ col_sum,
                        const float* __restrict__ diag, const float* __restrict__ mean_acc,
                        const float* __restrict__ tacc, float* __restrict__ out) {
  __shared__ float sh[256];
  int t = threadIdx.x;
  auto reduce = [&](float v) -> float {
    sh[t] = v; __syncthreads();
    for (int s = 128; s > 0; s >>= 1) { if (t < s) sh[t] += sh[t + s]; __syncthreads(); }
    float r = sh[0]; __syncthreads(); return r;
  };
  float s1 = 0.f, s2 = 0.f;
  for (int m = t; m < M_TOT; m += 256) {
    float d = diag[m];
    s1 += logf(row_sum[m]) - d;
    s2 += logf(col_sum[m]) - d;
  }
  float S1 = reduce(s1);
  float S2 = reduce(s2);
  float bq = 0.f;
  if (t < HIDN) { float v = mean_acc[t] / (float)M_TOT; bq = v * v; }
  float BQ = reduce(bq);
  if (t == 0) {
    float lc = 0.5f * (S1 / (float)M_TOT + S2 / (float)M_TOT);
    float nb = sqrtf(BQ);
    float ls = -BQ / fmaxf(nb, 1e-8f);
    float lt = tacc[0] / (4.f * (float)(L_SEQ - 1));
    out[0] = lc + ls + lt;
    out[1] = lc;
    out[2] = ls;
    out[3] = lt;
  }
}

extern "C" void kernel_launch(void* const* d_in, const int* in_sizes, int n_in,
                              void* d_out, int out_size, void* d_ws, size_t ws_size,
                              hipStream_t stream) {
  const float* h      = (const float*)d_in[0];
  const float* x_raw  = (const float*)d_in[1];
  const float* W_in   = (const float*)d_in[2];
  const float* b_in   = (const float*)d_in[3];
  const float* dt_emb = (const float*)d_in[4];
  const float* Wm1    = (const float*)d_in[5];
  const float* bm1    = (const float*)d_in[6];
  const float* Wm2    = (const float*)d_in[7];
  const float* bm2    = (const float*)d_in[8];
  const float* Wg     = (const float*)d_in[9];
  const float* a1     = (const float*)d_in[10];
  const float* a2     = (const float*)d_in[11];
  const float* Wf     = (const float*)d_in[12];
  const float* bf     = (const float*)d_in[13];
  float* ws = (float*)d_ws;
  float* out = (float*)d_out;

  k_zero<<<(ACC_N + 255) / 256, 256, 0, stream>>>(ws + OFF_ACC, ACC_N);
  k_gemm_zt  <<<128, 128, 0, stream>>>(h, W_in, b_in, ws + OFF_ZT);
  k_gemm_wh  <<<128, 128, 0, stream>>>(ws + OFF_ZT, Wg, ws + OFF_WH);
  k_gemm_sisj<<<128, 128, 0, stream>>>(x_raw, Wm1, ws + OFF_SISJ);
  k_rowdots  <<<512, 128, 0, stream>>>(ws + OFF_WH, a1, a2, ws + OFF_E1, ws + OFF_E2);
  k_attn     <<<512, 128, 0, stream>>>(ws + OFF_SISJ, dt_emb, Wm1, bm1, Wm2, bm2,
                                       ws + OFF_E1, ws + OFF_E2, ws + OFF_WH, ws + OFF_ZG);
  k_gemm_z   <<<128, 128, 0, stream>>>(ws + OFF_ZT, ws + OFF_ZG, Wf, bf, ws + OFF_Z);
  k_rowstats <<<512, 128, 0, stream>>>(ws + OFF_Z, ws + OFF_ZT, ws + OFF_ZG,
                                       ws + OFF_ZN, ws + OFF_QN, ws + OFF_KN, ws + OFF_MEAN);
  k_time     <<<512, 128, 0, stream>>>(ws + OFF_ZN, ws + OFF_TACC);
  k_logits   <<<4096, 128, 0, stream>>>(ws + OFF_QN, ws + OFF_KN,
                                        ws + OFF_RS, ws + OFF_CS, ws + OFF_DIAG);
  k_final    <<<1, 256, 0, stream>>>(ws + OFF_RS, ws + OFF_CS, ws + OFF_DIAG,
                                     ws + OFF_MEAN, ws + OFF_TACC, out);
}